// GINLayer_62380105007666
// MI455X (gfx1250) — compile-verified
//
#include <hip/hip_runtime.h>

#define N_NODES 100000
#define N_EDGES 1600000
#define D 128
#define BN_EPS 1e-5f
#define M_BLOCK 32
#define LDSS 132   // padded LDS row stride (floats): 132 % 64 == 4 -> conflict-free frag reads
#define SCAN_B 256
#define NB1 ((N_NODES + SCAN_B - 1) / SCAN_B)   // 391 scan blocks

typedef __attribute__((ext_vector_type(2))) float v2f;
typedef __attribute__((ext_vector_type(8))) float v8f;

// ---------------------------------------------------------------------------
// CSR build step 1: per-destination degree histogram (1.6M u32 atomics)
// ---------------------------------------------------------------------------
__global__ void k_count(const int* __restrict__ dst, unsigned* __restrict__ cnt) {
    unsigned e = blockIdx.x * blockDim.x + threadIdx.x;
    if (e < N_EDGES) atomicAdd(&cnt[dst[e]], 1u);
}

// ---------------------------------------------------------------------------
// CSR build step 2: exclusive scan of degrees (3 small kernels)
// ---------------------------------------------------------------------------
__global__ void k_scan1(const unsigned* __restrict__ cnt,
                        unsigned* __restrict__ off,
                        unsigned* __restrict__ bsum) {
    __shared__ unsigned s[SCAN_B];
    int t = threadIdx.x;
    int i = blockIdx.x * SCAN_B + t;
    unsigned v = (i < N_NODES) ? cnt[i] : 0u;
    s[t] = v;
    __syncthreads();
    #pragma unroll
    for (int d = 1; d < SCAN_B; d <<= 1) {
        unsigned a = (t >= d) ? s[t - d] : 0u;
        __syncthreads();
        s[t] += a;
        __syncthreads();
    }
    if (i < N_NODES) off[i] = s[t] - v;          // exclusive within block
    if (t == SCAN_B - 1) bsum[blockIdx.x] = s[t];
}

__global__ void k_scan2(unsigned* __restrict__ bsum) {
    __shared__ unsigned s[512];
    int t = threadIdx.x;
    unsigned v = (t < NB1) ? bsum[t] : 0u;
    s[t] = v;
    __syncthreads();
    #pragma unroll
    for (int d = 1; d < 512; d <<= 1) {
        unsigned a = (t >= d) ? s[t - d] : 0u;
        __syncthreads();
        s[t] += a;
        __syncthreads();
    }
    if (t < NB1) bsum[t] = s[t] - v;             // exclusive block offsets
}

__global__ void k_scan3(unsigned* __restrict__ off, const unsigned* __restrict__ bsum) {
    int i = blockIdx.x * SCAN_B + threadIdx.x;
    if (i < N_NODES) off[i] += bsum[blockIdx.x];
}

// ---------------------------------------------------------------------------
// CSR build step 3: drop each edge's src into its destination's slot list
// ---------------------------------------------------------------------------
__global__ void k_fill(const int* __restrict__ src, const int* __restrict__ dst,
                       const unsigned* __restrict__ off,
                       unsigned* __restrict__ cur, unsigned* __restrict__ csr) {
    unsigned e = blockIdx.x * blockDim.x + threadIdx.x;
    if (e >= N_EDGES) return;
    int d = dst[e];
    unsigned p = atomicAdd(&cur[d], 1u);
    csr[off[d] + p] = (unsigned)src[e];
}

// ---------------------------------------------------------------------------
// Aggregation as a gather (no f32 atomics):
// one wave per node, lane c owns floats [4c,4c+4); neighbor rows are
// coalesced 512B reads that hit the 192MB L2 (h table is 51.2MB).
// Fuses the (1+eps)*h self term.
// ---------------------------------------------------------------------------
__global__ void k_gather(const float* __restrict__ h,
                         const float* __restrict__ eps,
                         const unsigned* __restrict__ off,
                         const unsigned* __restrict__ cnt,
                         const unsigned* __restrict__ csr,
                         float* __restrict__ x) {
    unsigned gid = blockIdx.x * blockDim.x + threadIdx.x;
    unsigned node = gid >> 5;
    unsigned c = gid & 31;
    if (node >= N_NODES) return;
    unsigned o   = off[node];
    unsigned deg = cnt[node];
    float ax = 0.f, ay = 0.f, az = 0.f, aw = 0.f;
    for (unsigned j = 0; j < deg; ++j) {
        unsigned s = csr[o + j];                 // uniform across the wave
        float4 v = ((const float4*)(h + (size_t)s * D))[c];
        ax += v.x; ay += v.y; az += v.z; aw += v.w;
    }
    float se = 1.0f + eps[0];
    float4 hv = ((const float4*)(h + (size_t)node * D))[c];
    float4 r;
    r.x = se * hv.x + ax;
    r.y = se * hv.y + ay;
    r.z = se * hv.z + az;
    r.w = se * hv.w + aw;
    ((float4*)(x + (size_t)node * D))[c] = r;
}

// ---------------------------------------------------------------------------
// Fused 2-layer MLP with f32 WMMA + BN-stat accumulation (verified codegen).
// Block = 256 threads = 8 waves, 32 rows per block (100000 = 32 * 3125).
// ---------------------------------------------------------------------------
__global__ void __launch_bounds__(256) k_mlp(float* xbuf,
                                             const float* __restrict__ W1,
                                             const float* __restrict__ b1,
                                             const float* __restrict__ W2,
                                             const float* __restrict__ b2,
                                             float* gsum, float* gsq) {
    __shared__ float xs[M_BLOCK * LDSS];
    __shared__ float ssum[D];
    __shared__ float ssq[D];

    const int tid  = threadIdx.x;
    const int wave = tid >> 5;
    const int lane = tid & 31;
    const int row0 = blockIdx.x * M_BLOCK;

    if (tid < D) { ssum[tid] = 0.0f; ssq[tid] = 0.0f; }

    #pragma unroll
    for (int it = 0; it < 4; ++it) {
        int idx = tid + it * 256;
        int r = idx >> 5;
        int c = idx & 31;
        float4 v = ((const float4*)(xbuf + (size_t)(row0 + r) * D))[c];
        float* dp = &xs[r * LDSS + c * 4];
        dp[0] = v.x; dp[1] = v.y; dp[2] = v.z; dp[3] = v.w;
    }
    __syncthreads();

    const int nb    = wave * 16;
    const int nlane = lane & 15;
    const int khalf = (lane >> 4) * 2;
    const int mbase = (lane >> 4) * 8;

    v8f zero = {};
    v8f acc0 = zero, acc1 = zero;

    // ---- GEMM1: y1 = relu(x @ W1 + b1) ----
    #pragma unroll 4
    for (int kb = 0; kb < D; kb += 4) {
        int k = kb + khalf;
        v2f a0 = *(const v2f*)&xs[nlane * LDSS + k];
        v2f a1 = *(const v2f*)&xs[(16 + nlane) * LDSS + k];
        v2f b;
        b.x = W1[(size_t)k * D + nb + nlane];
        b.y = W1[(size_t)(k + 1) * D + nb + nlane];
        acc0 = __builtin_amdgcn_wmma_f32_16x16x4_f32(false, a0, false, b,
                                                     (short)0, acc0, false, false);
        acc1 = __builtin_amdgcn_wmma_f32_16x16x4_f32(false, a1, false, b,
                                                     (short)0, acc1, false, false);
    }
    __syncthreads();

    {
        float bias = b1[nb + nlane];
        #pragma unroll
        for (int v = 0; v < 8; ++v) {
            float y0 = acc0[v] + bias; y0 = y0 > 0.0f ? y0 : 0.0f;
            float y1 = acc1[v] + bias; y1 = y1 > 0.0f ? y1 : 0.0f;
            xs[(mbase + v) * LDSS + nb + nlane]      = y0;
            xs[(16 + mbase + v) * LDSS + nb + nlane] = y1;
        }
    }
    __syncthreads();

    // ---- GEMM2: y2 = y1 @ W2 + b2 ----
    acc0 = zero; acc1 = zero;
    #pragma unroll 4
    for (int kb = 0; kb < D; kb += 4) {
        int k = kb + khalf;
        v2f a0 = *(const v2f*)&xs[nlane * LDSS + k];
        v2f a1 = *(const v2f*)&xs[(16 + nlane) * LDSS + k];
        v2f b;
        b.x = W2[(size_t)k * D + nb + nlane];
        b.y = W2[(size_t)(k + 1) * D + nb + nlane];
        acc0 = __builtin_amdgcn_wmma_f32_16x16x4_f32(false, a0, false, b,
                                                     (short)0, acc0, false, false);
        acc1 = __builtin_amdgcn_wmma_f32_16x16x4_f32(false, a1, false, b,
                                                     (short)0, acc1, false, false);
    }

    {
        float bias = b2[nb + nlane];
        float s = 0.0f, q = 0.0f;
        #pragma unroll
        for (int v = 0; v < 8; ++v) {
            float y0 = acc0[v] + bias;
            float y1 = acc1[v] + bias;
            xbuf[(size_t)(row0 + mbase + v) * D + nb + nlane]      = y0;
            xbuf[(size_t)(row0 + 16 + mbase + v) * D + nb + nlane] = y1;
            s += y0 + y1;
            q += y0 * y0 + y1 * y1;
        }
        atomicAdd(&ssum[nb + nlane], s);
        atomicAdd(&ssq[nb + nlane], q);
    }
    __syncthreads();

    if (tid < D) {
        atomicAdd(&gsum[tid], ssum[tid]);
        atomicAdd(&gsq[tid], ssq[tid]);
    }
}

// ---------------------------------------------------------------------------
// Finalize BN scale/shift (one 128-thread block)
// ---------------------------------------------------------------------------
__global__ void k_bnstats(const float* __restrict__ gsum,
                          const float* __restrict__ gsq,
                          const float* __restrict__ gamma,
                          const float* __restrict__ beta,
                          float* __restrict__ sc, float* __restrict__ sh) {
    int j = threadIdx.x;
    float inv_n = 1.0f / (float)N_NODES;
    float mean  = gsum[j] * inv_n;
    float var   = gsq[j] * inv_n - mean * mean;
    float scale = gamma[j] * rsqrtf(var + BN_EPS);
    sc[j] = scale;
    sh[j] = beta[j] - mean * scale;
}

// ---------------------------------------------------------------------------
// out = h + relu(y2 * scale + shift)   (float4 streamed)
// ---------------------------------------------------------------------------
__global__ void k_epilogue(const float* __restrict__ h,
                           const float* __restrict__ xbuf,
                           const float* __restrict__ sc,
                           const float* __restrict__ sh,
                           float* __restrict__ out) {
    unsigned i = blockIdx.x * blockDim.x + threadIdx.x;
    const unsigned total = N_NODES * D / 4;
    if (i >= total) return;
    int c = (i & 31) * 4;
    float4 x  = ((const float4*)xbuf)[i];
    float4 hv = ((const float4*)h)[i];
    float4 a  = *(const float4*)&sc[c];
    float4 b  = *(const float4*)&sh[c];
    float4 r;
    r.x = hv.x + fmaxf(x.x * a.x + b.x, 0.0f);
    r.y = hv.y + fmaxf(x.y * a.y + b.y, 0.0f);
    r.z = hv.z + fmaxf(x.z * a.z + b.z, 0.0f);
    r.w = hv.w + fmaxf(x.w * a.w + b.w, 0.0f);
    ((float4*)out)[i] = r;
}

// ---------------------------------------------------------------------------
extern "C" void kernel_launch(void* const* d_in, const int* in_sizes, int n_in,
                              void* d_out, int out_size, void* d_ws, size_t ws_size,
                              hipStream_t stream) {
    const float* h     = (const float*)d_in[0];
    const int*   src   = (const int*)  d_in[1];
    const int*   dst   = (const int*)  d_in[2];
    const float* eps   = (const float*)d_in[3];
    const float* W1    = (const float*)d_in[4];
    const float* b1    = (const float*)d_in[5];
    const float* W2    = (const float*)d_in[6];
    const float* b2    = (const float*)d_in[7];
    const float* gamma = (const float*)d_in[8];
    const float* beta  = (const float*)d_in[9];
    float* out = (float*)d_out;

    // workspace layout
    float*    xbuf = (float*)d_ws;                         // N*D floats (51.2MB)
    unsigned* cnt  = (unsigned*)(xbuf + (size_t)N_NODES * D); // N u32 (degrees)
    unsigned* off  = cnt + N_NODES;                        // N u32 (offsets)
    unsigned* cur  = off + N_NODES;                        // N u32 (cursors)
    unsigned* bsum = cur + N_NODES;                        // 512 u32
    unsigned* csr  = bsum + 512;                           // E u32 (6.4MB)
    float*    gsum = (float*)(csr + N_EDGES);              // 128
    float*    gsq  = gsum + D;                             // 128
    float*    sc   = gsq + D;                              // 128
    float*    sh   = sc + D;                               // 128

    const unsigned elems4 = N_NODES * D / 4;               // 3.2M float4
    const unsigned eblk   = (N_EDGES + 255) / 256;         // 6250

    // --- CSR build ---
    hipMemsetAsync(cnt, 0, (size_t)N_NODES * sizeof(unsigned), stream);
    hipMemsetAsync(cur, 0, (size_t)N_NODES * sizeof(unsigned), stream);
    k_count<<<eblk, 256, 0, stream>>>(dst, cnt);
    k_scan1<<<NB1, SCAN_B, 0, stream>>>(cnt, off, bsum);
    k_scan2<<<1, 512, 0, stream>>>(bsum);
    k_scan3<<<NB1, SCAN_B, 0, stream>>>(off, bsum);
    k_fill<<<eblk, 256, 0, stream>>>(src, dst, off, cur, csr);

    // --- aggregation (gather) + eps self term ---
    k_gather<<<(N_NODES * 32) / 256, 256, 0, stream>>>(h, eps, off, cnt, csr, xbuf);

    // --- fused MLP (WMMA) + BN stats ---
    hipMemsetAsync(gsum, 0, 2 * D * sizeof(float), stream);
    k_mlp<<<N_NODES / M_BLOCK, 256, 0, stream>>>(xbuf, W1, b1, W2, b2, gsum, gsq);
    k_bnstats<<<1, D, 0, stream>>>(gsum, gsq, gamma, beta, sc, sh);
    k_epilogue<<<(elems4 + 255) / 256, 256, 0, stream>>>(h, xbuf, sc, sh, out);
}